// max_layer_77163382440516
// MI455X (gfx1250) — compile-verified
//
#include <hip/hip_runtime.h>
#include <hip/hip_bf16.h>

// Geometry from the reference
#define BATCH 256
#define SHIFTS 512
#define FILT 16
#define OUTF 32
#define NSCHUNK 8                    // s-chunks per batch-tile (grid.y)
#define SCHUNK (SHIFTS / NSCHUNK)    // 64 s per workgroup
#define SSTAGE 16                    // s rows staged in LDS per step
#define WPAD 36                      // padded o-stride -> conflict-free 16-lane b128 reads

typedef __attribute__((ext_vector_type(2))) float v2f;
typedef __attribute__((ext_vector_type(8))) float v8f;
typedef __attribute__((ext_vector_type(2))) float f32x2;
typedef __attribute__((ext_vector_type(4))) int   v4i;

#define AS3 __attribute__((address_space(3)))
#define AS1 __attribute__((address_space(1)))

// Order-preserving float -> u32 key (monotone: a < b  <=>  key(a) < key(b)).
// key(any finite float) > 0, so a memset-to-0 buffer acts as -inf.
__device__ __forceinline__ unsigned enc_key(float f) {
    int b = __float_as_int(f);
    unsigned ub = (unsigned)b;
    unsigned mask = (b < 0) ? 0xFFFFFFFFu : 0x80000000u;
    return ub ^ mask;
}
__device__ __forceinline__ float dec_key(unsigned k) {
    unsigned ub = (k & 0x80000000u) ? (k ^ 0x80000000u) : ~k;
    return __int_as_float((int)ub);
}

// ---------------------------------------------------------------------------
// Phase 1: per-(b,f,o) running max over a 64-shift chunk, merged with u32
// atomicMax on ordered keys.  256 threads = 16 local batches x 16 filters.
// W tile staged to LDS via CDNA5 async global->LDS DMA when available.
// ---------------------------------------------------------------------------
__global__ __launch_bounds__(256)
void partial_max_kernel(const float* __restrict__ xg,
                        const float* __restrict__ Wg,
                        unsigned* __restrict__ Mk) {
    __shared__ float sW[SSTAGE * FILT * WPAD];   // 16*16*36*4 = 36864 B

    const int t  = threadIdx.x;
    const int f  = t & 15;
    const int bl = t >> 4;
    const int b  = blockIdx.x * 16 + bl;
    const int sbase = blockIdx.y * SCHUNK;

    float m[OUTF];
#pragma unroll
    for (int o = 0; o < OUTF; ++o) m[o] = -INFINITY;

    for (int sc = 0; sc < SCHUNK / SSTAGE; ++sc) {
        const int s0 = sbase + sc * SSTAGE;

        __syncthreads();   // previous stage's LDS reads done (WAR on sW)

        // Stage W rows [s0, s0+16): 8192 floats = 2048 x b128, 8 per thread,
        // re-laid with WPAD stride for bank-conflict-free compute reads.
#pragma unroll
        for (int i = 0; i < 8; ++i) {
            const int idx = t + i * 256;     // b128 slot
            const int p   = idx << 2;        // flat float index in chunk
            const int s   = p >> 9;          // /512
            const int rem = p & 511;
            const int ff  = rem >> 5;
            const int oo  = rem & 31;        // multiple of 4
            const float* gsrc = Wg + (size_t)(s0 + s) * (FILT * OUTF) + rem;
            float* ldst = &sW[(s * FILT + ff) * WPAD + oo];
#if __has_builtin(__builtin_amdgcn_global_load_async_to_lds_b128)
            // Direct memory->LDS DMA (ASYNCcnt), bypassing VGPRs.
            __builtin_amdgcn_global_load_async_to_lds_b128(
                (AS1 v4i*)gsrc, (AS3 v4i*)ldst, 0, 0);
#else
            *reinterpret_cast<float4*>(ldst) =
                *reinterpret_cast<const float4*>(gsrc);
#endif
        }
#if __has_builtin(__builtin_amdgcn_global_load_async_to_lds_b128)
        asm volatile("s_wait_asynccnt 0x0" ::: "memory");
#endif
        // Speculative cache prefetch of the next W sub-tile (global_prefetch_b8)
        if (sc + 1 < SCHUNK / SSTAGE)
            __builtin_prefetch(Wg + (size_t)(s0 + SSTAGE) * (FILT * OUTF) +
                                   (size_t)t * 32, 0, 0);
        __syncthreads();   // LDS tile visible to whole workgroup

        for (int ss = 0; ss < SSTAGE; ++ss) {
            const float xv = xg[(size_t)b * (SHIFTS * FILT) +
                                (size_t)(s0 + ss) * FILT + f];
            const f32x2 xv2 = {xv, xv};
            const float* wr = &sW[(ss * FILT + f) * WPAD];
#pragma unroll
            for (int o4 = 0; o4 < OUTF / 4; ++o4) {
                const float4 wv = *reinterpret_cast<const float4*>(wr + (o4 << 2));
                f32x2 w01; w01[0] = wv.x; w01[1] = wv.y;
                f32x2 w23; w23[0] = wv.z; w23[1] = wv.w;
                // packed (v_pk_mul_f32) products; scalar/dual v_max_f32 maxima
                const f32x2 p01 = w01 * xv2;
                const f32x2 p23 = w23 * xv2;
                m[4 * o4 + 0] = fmaxf(m[4 * o4 + 0], p01[0]);
                m[4 * o4 + 1] = fmaxf(m[4 * o4 + 1], p01[1]);
                m[4 * o4 + 2] = fmaxf(m[4 * o4 + 2], p23[0]);
                m[4 * o4 + 3] = fmaxf(m[4 * o4 + 3], p23[1]);
            }
        }
    }

    unsigned* Mo = Mk + ((size_t)b * FILT + f) * OUTF;
#pragma unroll
    for (int o = 0; o < OUTF; ++o) atomicMax(Mo + o, enc_key(m[o]));
}

// ---------------------------------------------------------------------------
// Phase 2: sum over F=16 via v_wmma_f32_16x16x4_f32 with an all-ones B matrix
// (D[i,j] = rowsum(A_i) for every j), then bias + relu.
// Rows r = b*32 + o; one 16-row tile per wave; 4 K=4 WMMAs accumulate f=0..15.
// ---------------------------------------------------------------------------
__global__ __launch_bounds__(256)
void finalize_kernel(const unsigned* __restrict__ Mk,
                     const float* __restrict__ bias,
                     float* __restrict__ out) {
    const int lane = threadIdx.x & 31;
    const int wave = (int)((blockIdx.x * blockDim.x + threadIdx.x) >> 5); // 0..511
    const int r0   = wave << 4;              // tile base row (b constant in tile)
    const int row  = r0 + (lane & 15);       // this lane's A-matrix row
    const int b    = row >> 5;
    const int o    = row & 31;
    const int khalf = (lane >> 4) << 1;      // lanes 0-15 -> K{0,1}; 16-31 -> K{2,3}

    v2f ones; ones[0] = 1.0f; ones[1] = 1.0f;
    v8f acc = {};

    for (int c4 = 0; c4 < 4; ++c4) {
        const int k = c4 * 4 + khalf;        // filter index pair fed as K
        v2f a;
        a[0] = dec_key(Mk[((size_t)b * FILT + k) * OUTF + o]);
        a[1] = dec_key(Mk[((size_t)b * FILT + k + 1) * OUTF + o]);
        acc = __builtin_amdgcn_wmma_f32_16x16x4_f32(
            /*neg_a=*/false, a, /*neg_b=*/false, ones,
            /*c_mod=*/(short)0, acc, /*reuse_a=*/false, /*reuse_b=*/false);
    }

    // D layout: lane<16 holds M=v (v=0..7), lane>=16 holds M=v+8.
    // Lanes 0 and 16 each own 8 complete row-sums -> write 16 rows.
    if ((lane & 15) == 0) {
        const int mbase = (lane >> 4) * 8;
#pragma unroll
        for (int v = 0; v < 8; ++v) {
            const int r = r0 + mbase + v;
            const float val = acc[v] + bias[r & 31];
            out[r] = fmaxf(val, 0.0f);
        }
    }
}

// ---------------------------------------------------------------------------
extern "C" void kernel_launch(void* const* d_in, const int* in_sizes, int n_in,
                              void* d_out, int out_size, void* d_ws, size_t ws_size,
                              hipStream_t stream) {
    const float* x    = (const float*)d_in[0];   // [256, 8192]
    const float* W    = (const float*)d_in[1];   // [8192, 32]
    const float* bias = (const float*)d_in[2];   // [32]
    float* out        = (float*)d_out;           // [256, 32]
    unsigned* Mk      = (unsigned*)d_ws;         // 256*16*32 u32 keys = 512 KB

    // key(0x00000000) < key(any finite float)  ->  acts as -inf init
    (void)hipMemsetAsync(d_ws, 0, (size_t)BATCH * FILT * OUTF * sizeof(unsigned),
                         stream);

    dim3 gridB(BATCH / 16, NSCHUNK);
    partial_max_kernel<<<gridB, 256, 0, stream>>>(x, W, Mk);

    finalize_kernel<<<(BATCH * OUTF) / (16 * 8), 256, 0, stream>>>(Mk, bias, out);
}